// BAF_4105988735880
// MI455X (gfx1250) — compile-verified
//
#include <hip/hip_runtime.h>

// ---------------------------------------------------------------------------
// BAF (Block-Attention-Fusion) for MI455X / gfx1250, fp32 WMMA throughout.
// B=2, L=256, D=256, H=512, C=8, S=64, R=5
// ---------------------------------------------------------------------------

typedef __attribute__((ext_vector_type(2))) float v2f;
typedef __attribute__((ext_vector_type(8))) float v8f;

union V8 { v8f v; float f[8]; };

#define BB 2
#define LL 256
#define DD 256
#define HH 512
#define CC 8
#define SS 64
#define RR 5
#define CRS 2560   // C*S*R elems per (b,l) of a/bb
#define TRS 320    // S*R

// ---------------------------------------------------------------------------
// Generic strided + batched GEMM with bias:  O = A @ W + bias
// One wave computes a 16x64 strip of O (4 accumulators) via
// V_WMMA_F32_16X16X4_F32, reusing the A fragment across the 4 N-tiles.
// grid = (N/64, M/16, batch), block = 32.  Requires N % 64 == 0, K % 4 == 0.
// ---------------------------------------------------------------------------
__global__ void __launch_bounds__(32)
wmma_gemm_bias(const float* __restrict__ A, long aBatch, int lda,
               const float* __restrict__ W, long wBatch, int ldw,
               const float* __restrict__ bias, long biasBatch,
               float* __restrict__ O, long oBatch, int ldo,
               int M, int N, int K)
{
    const int lane = threadIdx.x;
    const int tn = blockIdx.x, tm = blockIdx.y, bz = blockIdx.z;
    const int hf  = lane >> 4;     // which half of the wave
    const int l15 = lane & 15;

    const float* Ab = A + (long)bz * aBatch;
    const float* Wb = W + (long)bz * wBatch;
    float*       Ob = O + (long)bz * oBatch;

    const int rowA = tm * 16 + l15;   // M index carried by this lane (A frag)
    const int col0 = tn * 64 + l15;   // first N index carried by this lane

    V8 acc[4];
    #pragma unroll
    for (int u = 0; u < 4; ++u)
        #pragma unroll
        for (int g = 0; g < 8; ++g) acc[u].f[g] = 0.0f;

    for (int k = 0; k < K; k += 4) {
        const int ka = k + hf * 2;   // K slots: lanes 0-15 -> k,k+1 ; 16-31 -> k+2,k+3
        v2f a;
        a.x = Ab[(long)rowA * lda + ka];
        a.y = Ab[(long)rowA * lda + ka + 1];
        const float* w0 = Wb + (long)ka * ldw;
        const float* w1 = Wb + (long)(ka + 1) * ldw;
        #pragma unroll
        for (int u = 0; u < 4; ++u) {
            v2f b;
            b.x = w0[col0 + u * 16];
            b.y = w1[col0 + u * 16];
            acc[u].v = __builtin_amdgcn_wmma_f32_16x16x4_f32(
                false, a, false, b, (short)0, acc[u].v, false, false);
        }
    }

    #pragma unroll
    for (int u = 0; u < 4; ++u) {
        const int col = col0 + u * 16;
        const float bv_ = bias ? (bias + (long)bz * biasBatch)[col] : 0.0f;
        #pragma unroll
        for (int g = 0; g < 8; ++g) {
            const int row = tm * 16 + g + hf * 8;   // C/D layout: VGPR g -> M=g (+8 hi)
            Ob[(long)row * ldo + col] = acc[u].f[g] + bv_;
        }
    }
}

// ---------------------------------------------------------------------------
// Block bilinear scores.
// scores[b,i,j] = bout + sum_c sum_s zn * Wout[c*64+s]
//   z[s]  = sum_{r<5} bbQ[b,i,c,r*64+s] * aK[b,j,c,r*64+s]  (two K=4 WMMAs, R->8)
//   y     = sign(z)*sqrt(|z|);  y^2 = |z|
//   norm  = sqrt(sum_s |z|)  -> pass 1 accumulates sum|z| only
//   zn    = y / max(norm, 1e-12)  -> pass 2 recomputes z from LDS
// Tiles are staged global->LDS with CDNA5 async-to-LDS (ASYNCcnt) B128 copies.
// grid = (L/16 (j), L/16 (i), B), block = 32 (one wave).
// ---------------------------------------------------------------------------
#define LDST 324   // padded row stride (dwords): 324*4 B keeps 16B alignment for
                   // B128 LDS writes; 324 % 64 == 4 -> 16 row-lanes hit 16 banks.

__global__ void __launch_bounds__(32)
block_scores_kernel(const float* __restrict__ bbQ,   // (B, L, CRS) query side
                    const float* __restrict__ aK,    // (B, L, CRS) key side
                    const float* __restrict__ Wout,  // (H)
                    const float* __restrict__ boutp, // (1)
                    float* __restrict__ scores)      // (B, L, L)
{
    __shared__ float ldsQ[16 * LDST];
    __shared__ float ldsK[16 * LDST];

    const int lane = threadIdx.x;
    const int tj = blockIdx.x, ti = blockIdx.y, b = blockIdx.z;
    const int hf  = lane >> 4;
    const int l15 = lane & 15;
    const int r0  = hf * 2;          // K slot base for this half

    float sc[8];
    #pragma unroll
    for (int g = 0; g < 8; ++g) sc[g] = 0.0f;

    const float* baseQ0 = bbQ + ((long)(b * LL + ti * 16)) * CRS;
    const float* baseK0 = aK  + ((long)(b * LL + tj * 16)) * CRS;

    for (int c = 0; c < CC; ++c) {
        // ---- async copy: 16 rows x 320 floats per side, B128 per lane ----
        // Previous chunk's ds reads must drain before async engine rewrites LDS.
        asm volatile("s_wait_dscnt 0x0" ::: "memory");
        const float* baseQ = baseQ0 + c * TRS;
        const float* baseK = baseK0 + c * TRS;
        #pragma unroll 1
        for (int it = 0; it < 40; ++it) {           // 40*32 lanes = 1280 float4/side
            const int e  = (it * 32 + lane) * 4;    // float index in 16x320 tile
            const int rr = e / TRS;
            const int t  = e % TRS;
            unsigned int       dQ = (unsigned int)(unsigned long)&ldsQ[rr * LDST + t];
            unsigned long long gQ = (unsigned long long)(baseQ + (long)rr * CRS + t);
            asm volatile("global_load_async_to_lds_b128 %0, %1, off"
                         :: "v"(dQ), "v"(gQ) : "memory");
            unsigned int       dK = (unsigned int)(unsigned long)&ldsK[rr * LDST + t];
            unsigned long long gK = (unsigned long long)(baseK + (long)rr * CRS + t);
            asm volatile("global_load_async_to_lds_b128 %0, %1, off"
                         :: "v"(dK), "v"(gK) : "memory");
        }
        asm volatile("s_wait_asynccnt 0x0" ::: "memory");
        __syncthreads();

        // ---- pass 1: accumulate sum_s |z| per (i,j) ----------------------
        float absacc[8];
        #pragma unroll
        for (int g = 0; g < 8; ++g) absacc[g] = 0.0f;

        for (int s = 0; s < SS; ++s) {
            V8 z;
            #pragma unroll
            for (int g = 0; g < 8; ++g) z.f[g] = 0.0f;
            v2f a0, b0, a1, b1;
            a0.x = ldsQ[l15 * LDST + r0 * SS + s];
            a0.y = ldsQ[l15 * LDST + (r0 + 1) * SS + s];
            b0.x = ldsK[l15 * LDST + r0 * SS + s];
            b0.y = ldsK[l15 * LDST + (r0 + 1) * SS + s];
            z.v = __builtin_amdgcn_wmma_f32_16x16x4_f32(false, a0, false, b0,
                                                        (short)0, z.v, false, false);
            const float qa = ldsQ[l15 * LDST + 4 * SS + s];   // r = 4
            const float ka = ldsK[l15 * LDST + 4 * SS + s];
            a1.x = hf ? 0.0f : qa;  a1.y = 0.0f;              // r = 5,6,7 padded zero
            b1.x = hf ? 0.0f : ka;  b1.y = 0.0f;
            z.v = __builtin_amdgcn_wmma_f32_16x16x4_f32(false, a1, false, b1,
                                                        (short)0, z.v, false, false);
            #pragma unroll
            for (int g = 0; g < 8; ++g) absacc[g] += fabsf(z.f[g]);
        }

        float rn[8];
        #pragma unroll
        for (int g = 0; g < 8; ++g)
            rn[g] = 1.0f / fmaxf(sqrtf(absacc[g]), 1e-12f);

        // ---- pass 2: recompute z, normalize, dot with Wout ---------------
        for (int s = 0; s < SS; ++s) {
            V8 z;
            #pragma unroll
            for (int g = 0; g < 8; ++g) z.f[g] = 0.0f;
            v2f a0, b0, a1, b1;
            a0.x = ldsQ[l15 * LDST + r0 * SS + s];
            a0.y = ldsQ[l15 * LDST + (r0 + 1) * SS + s];
            b0.x = ldsK[l15 * LDST + r0 * SS + s];
            b0.y = ldsK[l15 * LDST + (r0 + 1) * SS + s];
            z.v = __builtin_amdgcn_wmma_f32_16x16x4_f32(false, a0, false, b0,
                                                        (short)0, z.v, false, false);
            const float qa = ldsQ[l15 * LDST + 4 * SS + s];
            const float ka = ldsK[l15 * LDST + 4 * SS + s];
            a1.x = hf ? 0.0f : qa;  a1.y = 0.0f;
            b1.x = hf ? 0.0f : ka;  b1.y = 0.0f;
            z.v = __builtin_amdgcn_wmma_f32_16x16x4_f32(false, a1, false, b1,
                                                        (short)0, z.v, false, false);
            const float w = Wout[c * SS + s];
            #pragma unroll
            for (int g = 0; g < 8; ++g) {
                const float zz = z.f[g];
                const float y  = copysignf(sqrtf(fabsf(zz)), zz);
                sc[g] += y * rn[g] * w;
            }
        }
        __syncthreads();   // before next chunk overwrites LDS
    }

    const float bo = boutp[0];
    #pragma unroll
    for (int g = 0; g < 8; ++g) {
        const int i = ti * 16 + g + hf * 8;
        const int j = tj * 16 + l15;
        scores[((long)b * LL + i) * LL + j] = sc[g] + bo;
    }
}

// ---------------------------------------------------------------------------
// Softmax over last axis (j). One 256-thread block per (b,i) row, in place.
// ---------------------------------------------------------------------------
__global__ void __launch_bounds__(256)
softmax_kernel(float* __restrict__ scores)
{
    __shared__ float red[256];
    const int row = blockIdx.x;            // b*L + i
    const int t = threadIdx.x;
    const float x = scores[(long)row * LL + t];
    red[t] = x;
    __syncthreads();
    for (int off = 128; off > 0; off >>= 1) {
        if (t < off) red[t] = fmaxf(red[t], red[t + off]);
        __syncthreads();
    }
    const float m = red[0];
    __syncthreads();
    const float e = expf(x - m);
    red[t] = e;
    __syncthreads();
    for (int off = 128; off > 0; off >>= 1) {
        if (t < off) red[t] += red[t + off];
        __syncthreads();
    }
    scores[(long)row * LL + t] = e / red[0];
}

// ---------------------------------------------------------------------------
// Faithful swapaxes(1,2).reshape(B,-1,H):  perm[b,m,n] = atted[b, n%256, 2m + n/256]
// ---------------------------------------------------------------------------
__global__ void __launch_bounds__(256)
permute_kernel(const float* __restrict__ atted, float* __restrict__ perm)
{
    const int idx = blockIdx.x * 256 + threadIdx.x;   // < B*L*H = 262144
    const int n = idx & 511;
    const int m = (idx >> 9) & 255;
    const int b = idx >> 17;
    perm[idx] = atted[((long)b * LL + (n & 255)) * HH + 2 * m + (n >> 8)];
}

// ---------------------------------------------------------------------------
// BatchNorm1d (training stats over (b,d) per row i) + ReLU + residual.
// One 256-thread block per i; thread t covers d=t for b=0,1.
// ---------------------------------------------------------------------------
__global__ void __launch_bounds__(256)
bn_relu_res_kernel(const float* __restrict__ attn, const float* __restrict__ vin,
                   const float* __restrict__ gamma, const float* __restrict__ beta,
                   float* __restrict__ vout)
{
    __shared__ float s1[256], s2[256];
    const int i = blockIdx.x;
    const int t = threadIdx.x;
    const float x0 = attn[((long)0 * LL + i) * DD + t];
    const float x1 = attn[((long)1 * LL + i) * DD + t];
    s1[t] = x0 + x1;
    s2[t] = x0 * x0 + x1 * x1;
    __syncthreads();
    for (int off = 128; off > 0; off >>= 1) {
        if (t < off) { s1[t] += s1[t + off]; s2[t] += s2[t + off]; }
        __syncthreads();
    }
    const float mu   = s1[0] * (1.0f / 512.0f);
    const float var  = s2[0] * (1.0f / 512.0f) - mu * mu;
    const float rstd = rsqrtf(var + 1e-5f);
    const float g = gamma[i], be = beta[i];
    const float y0 = fmaxf((x0 - mu) * rstd * g + be, 0.0f) + vin[((long)0 * LL + i) * DD + t];
    const float y1 = fmaxf((x1 - mu) * rstd * g + be, 0.0f) + vin[((long)1 * LL + i) * DD + t];
    vout[((long)0 * LL + i) * DD + t] = y0;
    vout[((long)1 * LL + i) * DD + t] = y1;
}

__global__ void __launch_bounds__(256)
copy_kernel(const float* __restrict__ src, float* __restrict__ dst, int n)
{
    const int i = blockIdx.x * 256 + threadIdx.x;
    if (i < n) dst[i] = src[i];
}

// ---------------------------------------------------------------------------
extern "C" void kernel_launch(void* const* d_in, const int* in_sizes, int n_in,
                              void* d_out, int out_size, void* d_ws, size_t ws_size,
                              hipStream_t stream)
{
    const float* xi   = (const float*)d_in[0];
    const float* xt   = (const float*)d_in[1];
    const float* Wv   = (const float*)d_in[2];
    const float* bv   = (const float*)d_in[3];
    const float* Wk   = (const float*)d_in[4];
    const float* bk   = (const float*)d_in[5];
    const float* Wq   = (const float*)d_in[6];
    const float* bq   = (const float*)d_in[7];
    const float* Wm   = (const float*)d_in[8];
    const float* bm   = (const float*)d_in[9];
    const float* W0   = (const float*)d_in[10];
    const float* b0   = (const float*)d_in[11];
    const float* W1   = (const float*)d_in[12];
    const float* b1   = (const float*)d_in[13];
    const float* M0   = (const float*)d_in[14];
    const float* Mb0  = (const float*)d_in[15];
    const float* M1   = (const float*)d_in[16];
    const float* Mb1  = (const float*)d_in[17];
    const float* Wout = (const float*)d_in[18];
    const float* bout = (const float*)d_in[19];
    const float* gamma= (const float*)d_in[20];
    const float* beta = (const float*)d_in[21];

    // workspace carving (floats)
    float* ws     = (float*)d_ws;
    float* key_h  = ws;                  // B*L*H      = 262144
    float* x0k    = key_h  + 262144;     // B*L*H      = 262144
    float* aK     = x0k    + 262144;     // B*L*CRS    = 1310720
    float* vcur   = aK     + 1310720;    // B*L*D      = 131072
    float* value  = vcur   + 131072;     // B*L*H      = 262144
    float* query  = value  + 262144;     // B*L*H      = 262144
    float* x1     = query  + 262144;     // B*L*H      = 262144
    float* bbQ    = x1     + 262144;     // B*L*CRS    = 1310720
    float* scores = bbQ    + 1310720;    // B*L*L      = 131072
    float* atted  = scores + 131072;     // B*L*H      = 262144
    float* perm   = atted  + 262144;     // B*L*H      = 262144
    float* attn   = perm   + 262144;     // B*L*D      = 131072

    const dim3 w32(32);

    // ---- key-side precompute (depends only on xi; weights are shared) ----
    // key_h = xi @ Wk + bk            (M=512, K=256, N=512)
    wmma_gemm_bias<<<dim3(HH/64, (BB*LL)/16, 1), w32, 0, stream>>>(
        xi, 0, DD, Wk, 0, HH, bk, 0, key_h, 0, HH, BB*LL, HH, DD);
    // x0k = key_h @ W0 + b0           (512, 512, 512)
    wmma_gemm_bias<<<dim3(HH/64, (BB*LL)/16, 1), w32, 0, stream>>>(
        key_h, 0, HH, W0, 0, HH, b0, 0, x0k, 0, HH, BB*LL, HH, HH);
    // aK[b,l,c,:] = x0k[b,l,c,:] @ M0[c] + Mb0[c]   (batch c=8; M=512,K=64,N=320)
    wmma_gemm_bias<<<dim3(TRS/64, (BB*LL)/16, CC), w32, 0, stream>>>(
        x0k, SS, HH, M0, (long)SS*TRS, TRS, Mb0, TRS,
        aK, TRS, CRS, BB*LL, TRS, SS);

    // v = xt
    copy_kernel<<<dim3((BB*LL*DD)/256), dim3(256), 0, stream>>>(xt, vcur, BB*LL*DD);

    for (int layer = 0; layer < 2; ++layer) {
        // value = v @ Wv + bv ; query_h = v @ Wq + bq
        wmma_gemm_bias<<<dim3(HH/64, (BB*LL)/16, 1), w32, 0, stream>>>(
            vcur, 0, DD, Wv, 0, HH, bv, 0, value, 0, HH, BB*LL, HH, DD);
        wmma_gemm_bias<<<dim3(HH/64, (BB*LL)/16, 1), w32, 0, stream>>>(
            vcur, 0, DD, Wq, 0, HH, bq, 0, query, 0, HH, BB*LL, HH, DD);
        // x1 = query_h @ W1 + b1
        wmma_gemm_bias<<<dim3(HH/64, (BB*LL)/16, 1), w32, 0, stream>>>(
            query, 0, HH, W1, 0, HH, b1, 0, x1, 0, HH, BB*LL, HH, HH);
        // bbQ[b,l,c,:] = x1[b,l,c,:] @ M1[c] + Mb1[c]
        wmma_gemm_bias<<<dim3(TRS/64, (BB*LL)/16, CC), w32, 0, stream>>>(
            x1, SS, HH, M1, (long)SS*TRS, TRS, Mb1, TRS,
            bbQ, TRS, CRS, BB*LL, TRS, SS);

        // scores (B,L,L)
        block_scores_kernel<<<dim3(LL/16, LL/16, BB), w32, 0, stream>>>(
            bbQ, aK, Wout, bout, scores);
        // softmax over j, in place
        softmax_kernel<<<dim3(BB*LL), dim3(256), 0, stream>>>(scores);

        // atted = att @ value   (batch b; M=256, K=256, N=512)
        wmma_gemm_bias<<<dim3(HH/64, LL/16, BB), w32, 0, stream>>>(
            scores, (long)LL*LL, LL, value, (long)LL*HH, HH, nullptr, 0,
            atted, (long)LL*HH, HH, LL, HH, LL);

        // faithful transpose+view, then @ Wm + bm
        permute_kernel<<<dim3((BB*LL*HH)/256), dim3(256), 0, stream>>>(atted, perm);
        wmma_gemm_bias<<<dim3(DD/64, (BB*LL)/16, 1), w32, 0, stream>>>(
            perm, 0, HH, Wm, 0, DD, bm, 0, attn, 0, DD, BB*LL, DD, HH);

        // BatchNorm (stats per i over (b,d)) + ReLU + residual
        float* vdst = (layer == 0) ? vcur : (float*)d_out;
        bn_relu_res_kernel<<<dim3(LL), dim3(256), 0, stream>>>(
            attn, vcur, gamma, beta, vdst);
    }
}